// ContinuousAlphagrid_51032801411492
// MI455X (gfx1250) — compile-verified
//
#include <hip/hip_runtime.h>
#include <stdint.h>

// NeRF-style continuous alphagrid sampling for MI455X (gfx1250).
// Memory-bound streaming kernel: ~270MB traffic -> ~11.6us floor @ 23.3TB/s.
// No matmul structure -> no WMMA; CDNA5 win = async-to-LDS staging + NT stores.
// v2: branch-free neighbor read (v_cndmask instead of EXEC-masked LDS load),
//     redundant outer coordinate clamp removed.

typedef float v4f __attribute__((ext_vector_type(4)));

#define NB    16384   // rays
#define NSAMP 512     // samples per ray

__device__ __forceinline__ uint32_t expand_bits(uint32_t v) {
    v = (v * 65537u) & 4278190335u;
    v = (v * 257u)   & 251719695u;
    v = (v * 17u)    & 3272356035u;
    v = (v * 5u)     & 1227133513u;
    return v;
}

__global__ __launch_bounds__(256)
void alphagrid_sample_kernel(const float* __restrict__ rays_chunk,
                             const float* __restrict__ jitter,
                             const int*   __restrict__ bitfield,
                             float* __restrict__ out)
{
    constexpr float NEARV = 0.2f;
    constexpr float FARV  = 6.0f;
    constexpr float BOUND = 2.0f;
    constexpr int   GRIDN = 128;
    constexpr float STEP  = (FARV - NEARV) / (float)NSAMP;   // 0.011328125
    constexpr float HW    = STEP * 0.25f;                    // (t1-t0)/2, compile-time
    constexpr float HW2   = HW * HW;
    constexpr float HW4   = HW2 * HW2;
    constexpr float R2    = (0.0016f * 0.0016f) / 12.0f;     // BASE_RADIUS^2
    constexpr float INV_NORM = 1.0f / (BOUND - BOUND / (float)GRIDN);

    __shared__ float sj[2 * NSAMP];   // jitter tile: 2 rays x 512 samples = 4KB

    const int t    = threadIdx.x;          // 0..255
    const int ray0 = blockIdx.x * 2;

    // ---- Async DMA of the jitter tile into LDS (gfx1250 async VMEM path) ----
    {
        const float* gsrc = jitter + (size_t)ray0 * NSAMP + (size_t)t * 4;
        uint32_t lds_off  = (uint32_t)(uintptr_t)(&sj[t * 4]);   // low 32b of flat = LDS offset
        asm volatile("global_load_async_to_lds_b128 %0, %1, off"
                     :
                     : "v"(lds_off), "v"(gsrc)
                     : "memory");
    }
    asm volatile("s_wait_asynccnt 0" ::: "memory");
    __syncthreads();   // s_barrier_signal/-wait: publish LDS across the 8 waves

    // ---- Per-ray setup (wave-uniform-ish loads, fully cached) ----
    const int rloc = t >> 7;                 // 0..1: which ray in the block
    const int ray  = ray0 + rloc;
    const int s0   = (t & 127) * 4;          // first sample index handled by thread
    const bool has_next = (s0 + 4) < NSAMP;  // false only for the last thread of a ray

    const float* rc = rays_chunk + (size_t)ray * 6;
    const float ox = rc[0], oy = rc[1], oz = rc[2];
    const float dx = rc[3], dy = rc[4], dz = rc[5];
    const float dsx = dx * dx, dsy = dy * dy, dsz = dz * dz;
    const float dmag     = fmaxf(dsx + dsy + dsz, 1e-10f);
    const float inv_dmag = 1.0f / dmag;
    const float ndx = 1.0f - dsx * inv_dmag;
    const float ndy = 1.0f - dsy * inv_dmag;
    const float ndz = 1.0f - dsz * inv_dmag;

    // ---- Jitter from LDS: 16B read + one branch-free neighbor scalar ----
    const v4f jit4 = *(const v4f*)(&sj[rloc * NSAMP + s0]);
    // always-in-bounds wrapped read; value selected afterward (v_cndmask, no branch)
    const float jraw  = sj[rloc * NSAMP + ((s0 + 4) & (NSAMP - 1))];
    const float jnext = has_next ? jraw : 0.0f;

    float zv[5];
#pragma unroll
    for (int j = 0; j < 4; ++j)
        zv[j] = NEARV + STEP * ((float)(s0 + j) + jit4[j]);
    zv[4] = NEARV + STEP * ((float)(s0 + 4) + jnext);  // only meaningful when has_next

    // ---- Output layout (tuple concatenated flat, all stores NT 128-bit) ----
    const size_t base  = (size_t)ray * NSAMP + s0;
    float* xyz   = out;                                   // (B,NS,4)
    float* valid = out   + (size_t)NB * NSAMP * 4;        // (B,NS)
    float* zout  = valid + (size_t)NB * NSAMP;            // (B,NS)
    float* dout  = zout  + (size_t)NB * NSAMP;            // (B,NS)

    v4f zst, vst, dst_;

#pragma unroll
    for (int j = 0; j < 4; ++j) {
        const float z   = zv[j];
        const float mu  = z - 0.25f * STEP;
        const float mu2 = mu * mu;
        const float rd  = 1.0f / (3.0f * mu2 + HW2);

        const float t_mean = mu + 2.0f * mu * HW2 * rd;
        const float t_var  = HW2 * (1.0f / 3.0f)
                           - (4.0f / 15.0f) * (HW4 * (12.0f * mu2 - HW2)) * rd * rd;
        const float r_var  = R2 * (0.25f * mu2 + (5.0f / 12.0f) * HW2
                                   - (4.0f / 15.0f) * HW4 * rd);

        const float px = fmaf(dx, t_mean, ox);
        const float py = fmaf(dy, t_mean, oy);
        const float pz = fmaf(dz, t_mean, oz);

        const float vx = fmaf(t_var, dsx, r_var * ndx);
        const float vy = fmaf(t_var, dsy, r_var * ndy);
        const float vz = fmaf(t_var, dsz, r_var * ndz);
        const float maxv = fmaxf(vx, fmaxf(vy, vz));

        const bool outb = (fabsf(px) > BOUND) || (fabsf(py) > BOUND) || (fabsf(pz) > BOUND);

        // Morton-coded density bitfield lookup (1MB table, L2-resident).
        // Inner clamp to [-1,1] makes the outer [0,127] clip redundant in fp32:
        // (clamp(x,-1,1)+1)*63.5 is exactly within [0,127].
        const float fx = (fminf(fmaxf(px * INV_NORM, -1.0f), 1.0f) + 1.0f) * 63.5f;
        const float fy = (fminf(fmaxf(py * INV_NORM, -1.0f), 1.0f) + 1.0f) * 63.5f;
        const float fz = (fminf(fmaxf(pz * INV_NORM, -1.0f), 1.0f) + 1.0f) * 63.5f;
        uint32_t mort = expand_bits((uint32_t)fx)
                      | (expand_bits((uint32_t)fy) << 1)
                      | (expand_bits((uint32_t)fz) << 2);
        const uint32_t idx = (mort < 2097151u) ? mort : 2097151u;
        const uint32_t byte = (uint32_t)bitfield[idx >> 3];
        const bool alpha = (byte >> (idx & 7u)) & 1u;

        // xyz_sampled: one NT float4 store per sample (naturally 16B)
        v4f xo = { px, py, pz, maxv };
        __builtin_nontemporal_store(xo, (v4f*)(xyz + (base + (size_t)j) * 4));

        zst[j] = z;
        vst[j] = (!outb && alpha) ? 1.0f : 0.0f;

        // dists[i] = z[i+1]-z[i]; last sample of the ray -> 0 (branch-free select)
        const float znext = (j < 3) ? zv[j + 1] : (has_next ? zv[4] : z);
        dst_[j] = znext - z;
    }

    __builtin_nontemporal_store(zst,  (v4f*)(zout  + base));
    __builtin_nontemporal_store(vst,  (v4f*)(valid + base));
    __builtin_nontemporal_store(dst_, (v4f*)(dout  + base));
}

extern "C" void kernel_launch(void* const* d_in, const int* in_sizes, int n_in,
                              void* d_out, int out_size, void* d_ws, size_t ws_size,
                              hipStream_t stream) {
    (void)in_sizes; (void)n_in; (void)out_size; (void)d_ws; (void)ws_size;
    const float* rays_chunk = (const float*)d_in[0];
    const float* jitter     = (const float*)d_in[1];
    const int*   bitfield   = (const int*)d_in[2];
    float* out = (float*)d_out;

    dim3 grid(NB / 2);   // 8192 blocks, 2 rays each
    dim3 block(256);     // 8 wave32 per block
    hipLaunchKernelGGL(alphagrid_sample_kernel, grid, block, 0, stream,
                       rays_chunk, jitter, bitfield, out);
}